// GCN_28192165331202
// MI455X (gfx1250) — compile-verified
//
#include <hip/hip_runtime.h>
#include <hip/hip_bf16.h>

typedef __attribute__((ext_vector_type(2))) float v2f;
typedef __attribute__((ext_vector_type(8))) float v8f;

#define N_NODES 100000
#define N_EDGES 1600000
#define FDIM    128
#define OUT_DIM 10
#define BN_EPS  1e-5f
// Interleaved LDS layout for the 128x64 W panel: element (k,c) at word
// (k>>1)*PSTR + c*2 + (k&1). PSTR=160 -> row-pair stride mod 64 banks = 32,
// so the two half-waves read disjoint bank halves; each B frag is one ds_load_b64.
#define PSTR    160

// Fire-and-forget fp32 atomic add (lowers to global_atomic_add_f32, no return).
__device__ __forceinline__ void atomAddF(float* p, float v) {
    __hip_atomic_fetch_add(p, v, __ATOMIC_RELAXED, __HIP_MEMORY_SCOPE_AGENT);
}

// ---------------------------------------------------------------- utilities
__global__ void zero_kernel(float* __restrict__ p, long n) {
    long i = (long)blockIdx.x * blockDim.x + threadIdx.x;
    if (i < n) p[i] = 0.0f;
}

__global__ void deg_kernel(const long long* __restrict__ dst, float* __restrict__ deg, int e) {
    int i = blockIdx.x * blockDim.x + threadIdx.x;
    if (i < e) atomAddF(&deg[(int)dst[i]], 1.0f);
}

__global__ void dinv_kernel(float* __restrict__ deg, int n) {
    int i = blockIdx.x * blockDim.x + threadIdx.x;
    if (i < n) deg[i] = rsqrtf(deg[i] + 1.0f);
}

__global__ void norm_kernel(const long long* __restrict__ src, const long long* __restrict__ dst,
                            const float* __restrict__ dinv, float* __restrict__ nrm, int e) {
    int i = blockIdx.x * blockDim.x + threadIdx.x;
    if (i < e) nrm[i] = dinv[(int)src[i]] * dinv[(int)dst[i]];
}

// ---------------------------------------------------------------- hidden-layer WMMA GEMM
// H[N,128] = X[N,128] @ W[128,128]. Grid: (row-tile groups of 8, 2 column halves).
// Block: 256 threads = 8 waves; each wave computes a 16x64 strip (4 WMMA col-tiles)
// of its own 16-row tile, reusing one A fragment across 4 WMMAs per K-step.
// W panel staged in LDS with K-pair interleaving (see PSTR note above).
__global__ void gemm128_wmma(const float* __restrict__ X, const float* __restrict__ W,
                             float* __restrict__ H, int tilesM) {
    __shared__ __align__(16) float lw[64 * PSTR];   // 40 KB

    const int cg = blockIdx.y;          // column group: cols [cg*64, cg*64+64)
    // cooperative stage: 128 rows x 16 float4 each, scattered into interleaved layout
    for (int idx = threadIdx.x; idx < FDIM * 16; idx += 256) {
        int r  = idx >> 4;
        int c4 = (idx & 15) * 4;
        float4 w4 = *(const float4*)(W + r * FDIM + cg * 64 + c4);
        float* d = &lw[(r >> 1) * PSTR + (r & 1) + c4 * 2];
        d[0] = w4.x; d[2] = w4.y; d[4] = w4.z; d[6] = w4.w;
    }
    __syncthreads();

    const int wave = threadIdx.x >> 5;
    const int tm = blockIdx.x * 8 + wave;
    if (tm >= tilesM) return;           // wave-uniform (after the only barrier)

    const int lane = threadIdx.x & 31;
    const int lo = lane & 15;           // A row-in-tile / B,D col-in-tile
    const int hi = lane >> 4;           // K pair select within each K=4 step

    const float* xrow = X + (long)(tm * 16 + lo) * FDIM + 2 * hi;
    const v2f* lwv = (const v2f*)lw;    // float2 view: frag (k',c) at (k'>>1)*80 + c

    v8f a0 = {}, a1 = {}, a2 = {}, a3 = {};
#pragma unroll 4
    for (int k = 0; k < FDIM; k += 4) {
        v2f a;
        a.x = xrow[k];
        a.y = xrow[k + 1];
        const int rb = ((k >> 1) + hi) * (PSTR / 2) + lo;
        v2f b0 = lwv[rb];
        v2f b1 = lwv[rb + 16];
        v2f b2 = lwv[rb + 32];
        v2f b3 = lwv[rb + 48];
        a0 = __builtin_amdgcn_wmma_f32_16x16x4_f32(false, a, false, b0, (short)0, a0, false, false);
        a1 = __builtin_amdgcn_wmma_f32_16x16x4_f32(false, a, false, b1, (short)0, a1, false, false);
        a2 = __builtin_amdgcn_wmma_f32_16x16x4_f32(false, a, false, b2, (short)0, a2, false, false);
        a3 = __builtin_amdgcn_wmma_f32_16x16x4_f32(false, a, false, b3, (short)0, a3, false, false);
    }

    // D layout: VGPR r, lane(lo,hi) -> row tm*16 + r + 8*hi, col base + lo
    float* hb = H + (long)(tm * 16) * FDIM + cg * 64 + lo;
#pragma unroll
    for (int r = 0; r < 8; ++r) {
        long ro = (long)(r + 8 * hi) * FDIM;
        hb[ro]      = a0[r];
        hb[ro + 16] = a1[r];
        hb[ro + 32] = a2[r];
        hb[ro + 48] = a3[r];
    }
}

// ---------------------------------------------------------------- output-layer WMMA GEMM
// R[N,16] = X[N,128] @ W2[128,10] (cols 10..15 forced to zero, branchless mask).
__global__ void gemm_out_wmma(const float* __restrict__ X, const float* __restrict__ W2,
                              float* __restrict__ R, int tilesM) {
    int wave = (int)((blockIdx.x * blockDim.x + threadIdx.x) >> 5);
    if (wave >= tilesM) return;         // wave-uniform
    const int lane = threadIdx.x & 31;
    const int lo = lane & 15;
    const int hi = lane >> 4;

    const float* xrow = X + (long)(wave * 16 + lo) * FDIM + 2 * hi;
    const int cc = lo < OUT_DIM ? lo : OUT_DIM - 1;   // clamped column
    const float msk = lo < OUT_DIM ? 1.0f : 0.0f;     // v_cndmask, no EXEC change

    v8f acc = {};
#pragma unroll 8
    for (int k = 0; k < FDIM; k += 4) {
        v2f a;
        a.x = xrow[k];
        a.y = xrow[k + 1];
        const float* bp = W2 + (k + 2 * hi) * OUT_DIM + cc;
        v2f b;
        b.x = bp[0] * msk;
        b.y = bp[OUT_DIM] * msk;
        acc = __builtin_amdgcn_wmma_f32_16x16x4_f32(false, a, false, b, (short)0, acc, false, false);
    }

    float* rb = R + (long)(wave * 16) * 16 + lo;
#pragma unroll
    for (int r = 0; r < 8; ++r)
        rb[(r + 8 * hi) * 16] = acc[r];
}

// ---------------------------------------------------------------- aggregation
// A[dst] += norm[e] * H[src], 128-wide rows, float4 per lane, 32 lanes per edge
__global__ void agg128_kernel(const long long* __restrict__ src, const long long* __restrict__ dst,
                              const float* __restrict__ nrm, const float* __restrict__ H,
                              float* __restrict__ A, int e) {
    long t = (long)blockIdx.x * blockDim.x + threadIdx.x;
    long ee = t >> 5;
    if (ee >= e) return;
    int c = ((int)t & 31) * 4;
    int s = (int)src[ee];
    int d = (int)dst[ee];
    float w = nrm[ee];
    float4 v = *(const float4*)(H + (long)s * FDIM + c);
    float* ap = A + (long)d * FDIM + c;
    atomAddF(ap + 0, v.x * w);
    atomAddF(ap + 1, v.y * w);
    atomAddF(ap + 2, v.z * w);
    atomAddF(ap + 3, v.w * w);
}

// A += H * dinv^2 (self loop) + bias   over N x 128
__global__ void selfbias128_kernel(const float* __restrict__ H, const float* __restrict__ dinv,
                                   const float* __restrict__ b, float* __restrict__ A, long n128) {
    long i = (long)blockIdx.x * blockDim.x + threadIdx.x;
    if (i >= n128) return;
    long row = i >> 7;
    int c = (int)(i & 127);
    float di = dinv[row];
    A[i] += H[i] * di * di + b[c];
}

// ---------------------------------------------------------------- batch norm
__global__ void bnstats_kernel(const float* __restrict__ X, float* __restrict__ stats, int n) {
    int f = threadIdx.x;   // blockDim.x == 128
    float s = 0.0f, s2 = 0.0f;
    for (int r = blockIdx.x; r < n; r += gridDim.x) {
        float v = X[(long)r * FDIM + f];
        s += v;
        s2 += v * v;
    }
    atomAddF(&stats[f], s);
    atomAddF(&stats[FDIM + f], s2);
}

__global__ void bnfinal_kernel(float* __restrict__ stats) {
    int f = threadIdx.x;   // 128 threads
    float mean = stats[f] * (1.0f / N_NODES);
    float var  = stats[FDIM + f] * (1.0f / N_NODES) - mean * mean;
    stats[f] = mean;
    stats[FDIM + f] = rsqrtf(var + BN_EPS);
}

__global__ void bnapply_relu_kernel(const float* __restrict__ X, const float* __restrict__ g,
                                    const float* __restrict__ be, const float* __restrict__ stats,
                                    float* __restrict__ Y, long n128) {
    long i = (long)blockIdx.x * blockDim.x + threadIdx.x;
    if (i >= n128) return;
    int c = (int)(i & 127);
    float v = g[c] * (X[i] - stats[c]) * stats[FDIM + c] + be[c];
    Y[i] = v > 0.0f ? v : 0.0f;
}

// ---------------------------------------------------------------- final layer agg (10 cols, src padded to 16)
__global__ void agg10_kernel(const long long* __restrict__ src, const long long* __restrict__ dst,
                             const float* __restrict__ nrm, const float* __restrict__ R,
                             float* __restrict__ out, int e) {
    long t = (long)blockIdx.x * blockDim.x + threadIdx.x;
    long ee = t >> 4;
    int c = (int)t & 15;
    if (ee >= e || c >= OUT_DIM) return;
    int s = (int)src[ee];
    int d = (int)dst[ee];
    atomAddF(&out[(long)d * OUT_DIM + c], R[(long)s * 16 + c] * nrm[ee]);
}

__global__ void selfbias10_kernel(const float* __restrict__ R, const float* __restrict__ dinv,
                                  const float* __restrict__ b, float* __restrict__ out, long n10) {
    long i = (long)blockIdx.x * blockDim.x + threadIdx.x;
    if (i >= n10) return;
    long row = i / OUT_DIM;
    int c = (int)(i % OUT_DIM);
    float di = dinv[row];
    out[i] += R[row * 16 + c] * di * di + b[c];
}

// ---------------------------------------------------------------- launch
extern "C" void kernel_launch(void* const* d_in, const int* in_sizes, int n_in,
                              void* d_out, int out_size, void* d_ws, size_t ws_size,
                              hipStream_t stream) {
    const float*     x   = (const float*)d_in[0];
    const long long* ei  = (const long long*)d_in[1];
    const long long* src = ei;                 // edge_index[0]
    const long long* dst = ei + N_EDGES;       // edge_index[1]
    const float* W0 = (const float*)d_in[2];
    const float* b0 = (const float*)d_in[3];
    const float* g0 = (const float*)d_in[4];
    const float* be0 = (const float*)d_in[5];
    const float* W1 = (const float*)d_in[6];
    const float* b1 = (const float*)d_in[7];
    const float* g1 = (const float*)d_in[8];
    const float* be1 = (const float*)d_in[9];
    const float* W2 = (const float*)d_in[10];
    const float* b2 = (const float*)d_in[11];
    float* out = (float*)d_out;

    float* ws    = (float*)d_ws;
    float* dinv  = ws;                                  // N
    float* nrm   = dinv + N_NODES;                      // E
    float* P     = nrm + N_EDGES;                       // N*128
    float* Q     = P + (long)N_NODES * FDIM;            // N*128
    float* R     = Q + (long)N_NODES * FDIM;            // N*16 (padded layer-2 out)
    float* stats = R + (long)N_NODES * 16;              // 256

    const int T = 256;
    const long NF = (long)N_NODES * FDIM;
    const int gNF = (int)((NF + T - 1) / T);
    const int gE  = (N_EDGES + T - 1) / T;
    const int gN  = (N_NODES + T - 1) / T;
    const int tilesM = N_NODES / 16;                    // 6250 exactly
    const int gAgg128 = (int)(((long)N_EDGES * 32 + T - 1) / T);
    const int gAgg10  = (int)(((long)N_EDGES * 16 + T - 1) / T);
    const dim3 gGemm((tilesM + 7) / 8, 2);              // 782 x 2 column halves

    // degree -> dinv -> per-edge norm (shared by all layers)
    zero_kernel<<<gN, T, 0, stream>>>(dinv, N_NODES);
    deg_kernel<<<gE, T, 0, stream>>>(dst, dinv, N_EDGES);
    dinv_kernel<<<gN, T, 0, stream>>>(dinv, N_NODES);
    norm_kernel<<<gE, T, 0, stream>>>(src, dst, dinv, nrm, N_EDGES);

    // ---- layer 0: P = x@W0 ; Q = aggregate(P) + self + b0 ; P = relu(BN(Q))
    gemm128_wmma<<<gGemm, T, 0, stream>>>(x, W0, P, tilesM);
    zero_kernel<<<gNF, T, 0, stream>>>(Q, NF);
    agg128_kernel<<<gAgg128, T, 0, stream>>>(src, dst, nrm, P, Q, N_EDGES);
    selfbias128_kernel<<<gNF, T, 0, stream>>>(P, dinv, b0, Q, NF);
    zero_kernel<<<1, 256, 0, stream>>>(stats, 256);
    bnstats_kernel<<<512, 128, 0, stream>>>(Q, stats, N_NODES);
    bnfinal_kernel<<<1, 128, 0, stream>>>(stats);
    bnapply_relu_kernel<<<gNF, T, 0, stream>>>(Q, g0, be0, stats, P, NF);

    // ---- layer 1: Q = P@W1 ; P = aggregate(Q) + self + b1 ; Q = relu(BN(P))
    gemm128_wmma<<<gGemm, T, 0, stream>>>(P, W1, Q, tilesM);
    zero_kernel<<<gNF, T, 0, stream>>>(P, NF);
    agg128_kernel<<<gAgg128, T, 0, stream>>>(src, dst, nrm, Q, P, N_EDGES);
    selfbias128_kernel<<<gNF, T, 0, stream>>>(Q, dinv, b1, P, NF);
    zero_kernel<<<1, 256, 0, stream>>>(stats, 256);
    bnstats_kernel<<<512, 128, 0, stream>>>(P, stats, N_NODES);
    bnfinal_kernel<<<1, 128, 0, stream>>>(stats);
    bnapply_relu_kernel<<<gNF, T, 0, stream>>>(P, g1, be1, stats, Q, NF);

    // ---- layer 2: R = Q@W2 (10 cols padded to 16) ; out = aggregate(R) + self + b2
    gemm_out_wmma<<<(tilesM + 7) / 8, T, 0, stream>>>(Q, W2, R, tilesM);
    zero_kernel<<<(int)(((long)N_NODES * OUT_DIM + T - 1) / T), T, 0, stream>>>(out, (long)N_NODES * OUT_DIM);
    agg10_kernel<<<gAgg10, T, 0, stream>>>(src, dst, nrm, R, out, N_EDGES);
    selfbias10_kernel<<<(int)(((long)N_NODES * OUT_DIM + T - 1) / T), T, 0, stream>>>(
        R, dinv, b2, out, (long)N_NODES * OUT_DIM);
}